// warp3D_13374528160334
// MI455X (gfx1250) — compile-verified
//
#include <hip/hip_runtime.h>

// 3D trilinear warp (backward warp with flow), matching the JAX reference
// (weights use the CLAMPED upper corner; trilinear sum evaluated in exactly
// factored lerp form — tensor-product equivalent of the 8-weight sum).
//
// Shapes: I[2,2,160,192,224], flow[2,3,160,192,224], out[2,2,160,192,224] f32.
//
// ~385 MB traffic @ 23.3 TB/s => ~16.5 us floor. The naive kernel is
// instruction-issue bound, so this version minimizes instructions/point:
//   - 4 output points per thread (2x2 tile in y,z): wave-setup SALU and the
//     64-bit flow/store base construction amortize 4x; the constant sp deltas
//     (W, HW, HW+W elements) fold into 24-bit instruction offsets.
//   - x-corner pairs fused into single b64 gathers; x-clamp folded into
//     per-point x-weights (wl, wh) — no per-pair value selects.
//   - factored trilinear lerp (no 8-weight expansion).
//   - flow: NT loads (streamed once); out: NT stores (write once);
//     I: RT gathers (110 MB fits in 192 MB L2, ~8x reuse).
//   - 3D launch grid does the (b,z,y,x) decomposition; no div, no bounds check.

#define B_   2
#define C_   2
#define D_   160
#define H_   192
#define W_   224
#define HW_  (H_ * W_)        // 43008
#define DHW_ (D_ * H_ * W_)   // 6881280

__device__ __forceinline__ float2 ldpair(const float* __restrict__ p) {
    float2 v;
    __builtin_memcpy(&v, p, sizeof(float2));   // b64 load, 4B-aligned OK
    return v;
}

__global__ __launch_bounds__(W_) void warp3d_trilinear_kernel(
    const float* __restrict__ I,
    const float* __restrict__ flow,
    float* __restrict__ out)
{
    const unsigned xi = threadIdx.x;        // 0..223
    const unsigned yb = blockIdx.x * 2u;    // 0..190 step 2
    const unsigned zb = blockIdx.y * 2u;    // 0..158 step 2
    const unsigned b  = blockIdx.z;         // 0..1

    // Wave-uniform bases (built once; channel 1 = channel 0 + DHW).
    const float* __restrict__ fb  = flow + (size_t)b * (3u * DHW_);
    const float* __restrict__ Iv0 = I    + (size_t)(b * C_) * DHW_;
    const float* __restrict__ Iv1 = Iv0 + DHW_;
    float*       __restrict__ ov0 = out  + (size_t)(b * C_) * DHW_;
    float*       __restrict__ ov1 = ov0 + DHW_;

    const float Xf = (float)xi;   // X coordinate base is the same for all 4 pts

#pragma unroll
    for (unsigned pz = 0; pz < 2; ++pz) {
#pragma unroll
        for (unsigned py = 0; py < 2; ++py) {
            const unsigned yi = yb + py;
            const unsigned zi = zb + pz;
            const unsigned sp = (zi * H_ + yi) * W_ + xi;

            // ---- flow reads: shared base, constant deltas -> ioffset ----
            const float fx = __builtin_nontemporal_load(fb + sp);
            const float fy = __builtin_nontemporal_load(fb + sp + DHW_);
            const float fz = __builtin_nontemporal_load(fb + sp + 2u * DHW_);

            const float X = fx + Xf;
            const float Y = fy + (float)yi;
            const float Z = fz + (float)zi;

            const int x0s = (int)floorf(X);
            const int y0s = (int)floorf(Y);
            const int z0s = (int)floorf(Z);

            // Clamped corner indices (reference semantics).
            const int x1c = min(max(x0s + 1, 0), W_ - 1);
            const int y1c = min(max(y0s + 1, 0), H_ - 1);
            const int z1c = min(max(z0s + 1, 0), D_ - 1);
            const int y0c = min(max(y0s, 0), H_ - 1);
            const int z0c = min(max(z0s, 0), D_ - 1);

            // Fractions use the CLAMPED upper corner (reference quirk).
            const float dx = (float)x1c - X;
            const float dy = (float)y1c - Y;
            const float dz = (float)z1c - Z;
            const float ex = 1.0f - dx;
            const float ey = 1.0f - dy;
            const float ez = 1.0f - dz;

            // Fused b64 x-pair at x0p = clamp(x0s, 0, W-2); x-clamp folded
            // into (wl, wh).
            const unsigned x0p = (unsigned)min(max(x0s, 0), W_ - 2);
            const bool xlo = (x0s < 0);        // both corners -> col 0 (pair.x)
            const bool xhi = (x0s > W_ - 2);   // both corners -> col W-1 (pair.y)
            const float sw = dx + ex;
            const float wl = xlo ? sw : (xhi ? 0.0f : dx);
            const float wh = xhi ? sw : (xlo ? 0.0f : ex);

            // Row base offsets (include pair base), shared across channels.
            const unsigned r00 = (unsigned)z0c * HW_ + (unsigned)y0c * W_ + x0p;
            const unsigned r01 = (unsigned)z0c * HW_ + (unsigned)y1c * W_ + x0p;
            const unsigned r10 = (unsigned)z1c * HW_ + (unsigned)y0c * W_ + x0p;
            const unsigned r11 = (unsigned)z1c * HW_ + (unsigned)y1c * W_ + x0p;

            // ---- channel 0 ----
            {
                const float2 p00 = ldpair(Iv0 + r00);
                const float2 p01 = ldpair(Iv0 + r01);
                const float2 p10 = ldpair(Iv0 + r10);
                const float2 p11 = ldpair(Iv0 + r11);

                const float rx00 = fmaf(wl, p00.x, wh * p00.y);
                const float rx01 = fmaf(wl, p01.x, wh * p01.y);
                const float rx10 = fmaf(wl, p10.x, wh * p10.y);
                const float rx11 = fmaf(wl, p11.x, wh * p11.y);

                const float gy0 = fmaf(dy, rx00, ey * rx01);
                const float gy1 = fmaf(dy, rx10, ey * rx11);
                __builtin_nontemporal_store(fmaf(dz, gy0, ez * gy1), ov0 + sp);
            }
            // ---- channel 1 ----
            {
                const float2 p00 = ldpair(Iv1 + r00);
                const float2 p01 = ldpair(Iv1 + r01);
                const float2 p10 = ldpair(Iv1 + r10);
                const float2 p11 = ldpair(Iv1 + r11);

                const float rx00 = fmaf(wl, p00.x, wh * p00.y);
                const float rx01 = fmaf(wl, p01.x, wh * p01.y);
                const float rx10 = fmaf(wl, p10.x, wh * p10.y);
                const float rx11 = fmaf(wl, p11.x, wh * p11.y);

                const float gy0 = fmaf(dy, rx00, ey * rx01);
                const float gy1 = fmaf(dy, rx10, ey * rx11);
                __builtin_nontemporal_store(fmaf(dz, gy0, ez * gy1), ov1 + sp);
            }
        }
    }
}

extern "C" void kernel_launch(void* const* d_in, const int* in_sizes, int n_in,
                              void* d_out, int out_size, void* d_ws, size_t ws_size,
                              hipStream_t stream) {
    (void)in_sizes; (void)n_in; (void)out_size; (void)d_ws; (void)ws_size;
    const float* I    = (const float*)d_in[0];   // [B,C,D,H,W]
    const float* flow = (const float*)d_in[1];   // [B,3,D,H,W]
    float* out        = (float*)d_out;           // [B,C,D,H,W]

    dim3 block(W_, 1, 1);             // 224 threads = 7 wave32s
    dim3 grid(H_ / 2, D_ / 2, B_);    // 2x2 (y,z) tile per thread
    warp3d_trilinear_kernel<<<grid, block, 0, stream>>>(I, flow, out);
}